// SparseLinear_27504970563839
// MI455X (gfx1250) — compile-verified
//
#include <hip/hip_runtime.h>
#include <stdint.h>

// Problem constants (fixed by the reference).
constexpr int B     = 2048;
constexpr int N_IN  = 8192;
constexpr int N_OUT = 8192;
constexpr int K     = 9;

// Tiling.
constexpr int THREADS    = 256;            // 8 waves (wave32)
constexpr int WAVES      = THREADS / 32;
constexpr int B_TILE     = 8;              // x rows staged in LDS per workgroup (= WAVES)
constexpr int ROW_PAD    = 8;              // floats; gather banks (c + 8r) % 64 distinct, r=0..7
constexpr int ROW_STRIDE = N_IN + ROW_PAD; // 8200 floats
constexpr int O_TILES    = 4;              // grid.x -> x re-read from L2 only 4x (256 MiB)
constexpr int O_TILE     = N_OUT / O_TILES;     // 2048 outputs / workgroup
constexpr int O_PER_WAVE = O_TILE / WAVES;      // 256 outputs / wave
constexpr int O_SUBS     = 4;                   // output subgroups per wave (lane>>3)
constexpr int STEPS      = O_PER_WAVE / O_SUBS; // 64 consecutive outputs per lane

constexpr size_t LDS_BYTES = (size_t)B_TILE * ROW_STRIDE * sizeof(float); // 262,400 B

// One row = 8192 floats = 2048 x b128 chunks = 64 wave-instructions x 512 B.
constexpr int STAGE_ITERS  = (N_IN * 4) / 512;  // 64
constexpr int STAGE_STRIDE = 512;               // bytes per wave-instruction

// ---- CDNA5 async global->LDS helpers --------------------------------------

// 32-bit LDS byte offset of a generic pointer into shared memory.
static __device__ __forceinline__ uint32_t lds_off_u32(const void* p) {
    return (uint32_t)(uintptr_t)(__attribute__((address_space(3))) const void*)p;
}

// Fully unrolled async row stage: base addresses live in 3 VGPRs, each issue
// uses only the 24-bit instruction immediate (applies to BOTH global and LDS
// addresses per the ISA pseudocode) -> zero per-iteration VALU.
template <int IT, int N>
struct StageUnroll {
    static __device__ __forceinline__ void run(uint32_t lds_off, uint64_t gaddr) {
        asm volatile("global_load_async_to_lds_b128 %0, %1, off offset:%2"
                     :: "v"(lds_off), "v"(gaddr), "i"(IT * STAGE_STRIDE)
                     : "memory");
        StageUnroll<IT + 1, N>::run(lds_off, gaddr);
    }
};
template <int N>
struct StageUnroll<N, N> {
    static __device__ __forceinline__ void run(uint32_t, uint64_t) {}
};

static __device__ __forceinline__ void wait_asynccnt_zero() {
#if __has_builtin(__builtin_amdgcn_s_wait_asynccnt)
    __builtin_amdgcn_s_wait_asynccnt(0);
#else
    asm volatile("s_wait_asynccnt 0" ::: "memory");
#endif
}

// ---- Kernel ---------------------------------------------------------------

__global__ __launch_bounds__(THREADS, 1)
void sparse_linear_gather_kernel(const float* __restrict__ x,
                                 const int*   __restrict__ idx,
                                 const float* __restrict__ vals,
                                 const float* __restrict__ bias,
                                 float*       __restrict__ out) {
    extern __shared__ float lds_x[];   // [B_TILE][ROW_STRIDE], padded rows

    const int tid  = (int)threadIdx.x;
    const int wave = tid >> 5;
    const int lane = tid & 31;
    const int b0   = (int)blockIdx.y * B_TILE;     // first batch row of tile
    const int o_wg = (int)blockIdx.x * O_TILE;     // first output of tile

    // ---- Stage B_TILE rows of x into LDS (CDNA5 async DMA, ASYNCcnt) ----
    // Wave w stages row w: 64 unrolled b128 asyncs, immediate offsets only.
    {
        const float*  src      = x + (size_t)(b0 + wave) * N_IN + (size_t)lane * 4;
        const uint32_t dst_lds = lds_off_u32(&lds_x[wave * ROW_STRIDE]) + lane * 16;
        StageUnroll<0, STAGE_ITERS>::run(dst_lds, (uint64_t)(uintptr_t)src);
        wait_asynccnt_zero();   // each wave drains its own ASYNCcnt...
    }
    __syncthreads();            // ...then the barrier makes all rows visible.

    // ---- Gather + accumulate ----
    // lane = 4 output-subgroups x 8 batch rows. The 8 lanes of a subgroup read
    // the SAME column c from 8 padded rows: banks (c + 8r) mod 64 are distinct,
    // so LDS gathers are (near) conflict-free. Each lane owns STEPS consecutive
    // outputs -> float4 result stores.
    const int o_sub = lane >> 3;           // 0..3
    const int r     = lane & 7;            // batch row within tile

    const float* __restrict__ lds_row = lds_x + r * ROW_STRIDE;
    const int o_first = o_wg + wave * O_PER_WAVE + o_sub * STEPS;  // multiple of 4
    float* __restrict__ out_base = out + (size_t)(b0 + r) * N_OUT + o_first;

    for (int q4 = 0; q4 < STEPS; q4 += 4) {
        float acc[4];
        #pragma unroll
        for (int j = 0; j < 4; ++j) {
            const int o = o_first + q4 + j;
            const int*   __restrict__ ip = idx  + (size_t)o * K;
            const float* __restrict__ vp = vals + (size_t)o * K;

            int   c[K];
            float v[K];
            #pragma unroll
            for (int k = 0; k < K; ++k) { c[k] = ip[k]; v[k] = vp[k]; }

            float a = bias[o];
            #pragma unroll
            for (int k = 0; k < K; ++k) {
                a = fmaf(lds_row[c[k]], v[k], a);   // ds_load_b32 + v_fmac_f32
            }
            acc[j] = a;
        }
        // 16B-aligned: o_first % 4 == 0 and q4 % 4 == 0 -> global_store_b128.
        float4 res; res.x = acc[0]; res.y = acc[1]; res.z = acc[2]; res.w = acc[3];
        *reinterpret_cast<float4*>(out_base + q4) = res;
    }
}

// ---- Host launch ----------------------------------------------------------

extern "C" void kernel_launch(void* const* d_in, const int* in_sizes, int n_in,
                              void* d_out, int out_size, void* d_ws, size_t ws_size,
                              hipStream_t stream) {
    (void)in_sizes; (void)n_in; (void)out_size; (void)d_ws; (void)ws_size;

    const float* x    = (const float*)d_in[0];
    const int*   idx  = (const int*)  d_in[1];
    const float* vals = (const float*)d_in[2];
    const float* bias = (const float*)d_in[3];
    float*       out  = (float*)d_out;

    // Allow >64KB dynamic LDS (gfx1250 supports up to 320KB per workgroup).
    (void)hipFuncSetAttribute((const void*)sparse_linear_gather_kernel,
                              hipFuncAttributeMaxDynamicSharedMemorySize,
                              (int)LDS_BYTES);

    dim3 grid(O_TILES, B / B_TILE);   // 4 x 256 = 1024 workgroups
    sparse_linear_gather_kernel<<<grid, THREADS, LDS_BYTES, stream>>>(
        x, idx, vals, bias, out);
}